// FrozenInputToLeaf_48670569398603
// MI455X (gfx1250) — compile-verified
//
#include <hip/hip_runtime.h>
#include <stdint.h>

// out[b, leaf] = x[b, idx[leaf]]  -- pure bandwidth problem (~384 MB -> ~16.5us @ 23.3 TB/s).
// Strategy: async-DMA 8 rows of x into LDS (CDNA5 GLOBAL_LOAD_ASYNC_TO_LDS_B128,
// ASYNCcnt-tracked), gather from LDS (random 4B access is cheap there), stream
// results out with coalesced non-temporal 16B stores.

typedef float v4f __attribute__((ext_vector_type(4)));
typedef int   v4i __attribute__((ext_vector_type(4)));

#define NUM_INPUTS_C 4096
#define NUM_LEAVES_C 2048
#define ROWS_PER_BLOCK 8
#define THREADS 256   // 8 wave32s

__global__ __launch_bounds__(THREADS)
void gather_cols_kernel(const float* __restrict__ x,
                        const int* __restrict__ idx,
                        float* __restrict__ out) {
  __shared__ float tile[ROWS_PER_BLOCK * NUM_INPUTS_C];  // 128 KB of the 320 KB WGP LDS

  const int tid = threadIdx.x;
  const size_t row0 = (size_t)blockIdx.x * ROWS_PER_BLOCK;
  const float* __restrict__ src = x + row0 * NUM_INPUTS_C;

  // ---- Phase 1: stage ROWS_PER_BLOCK rows of x into LDS (16 B per lane per issue) ----
  constexpr int TOTAL_V4 = ROWS_PER_BLOCK * NUM_INPUTS_C / 4;  // 8192 x float4
  constexpr int ITERS    = TOTAL_V4 / THREADS;                 // 32 per thread

#if __has_builtin(__builtin_amdgcn_global_load_async_to_lds_b128)
  {
    const int e0 = tid * 4;  // float offset of this lane's first 16B chunk
#pragma unroll
    for (int i = 0; i < ITERS; ++i) {
      const int e = e0 + i * (THREADS * 4);
      __builtin_amdgcn_global_load_async_to_lds_b128(
          (v4i*)(src + e), (v4i*)(tile + e), /*offset=*/0, /*cpol=*/0);
    }
#if __has_builtin(__builtin_amdgcn_s_wait_asynccnt)
    __builtin_amdgcn_s_wait_asynccnt(0);
#else
    asm volatile("s_wait_asynccnt 0" ::: "memory");
#endif
  }
#else
  {
    // Fallback: coalesced b128 loads through VGPRs + ds_store_b128.
    const v4f* __restrict__ src4  = (const v4f*)src;
    v4f* __restrict__ tile4 = (v4f*)tile;
#pragma unroll
    for (int i = 0; i < ITERS; ++i) {
      const int e = tid + i * THREADS;
      tile4[e] = src4[e];
    }
  }
#endif
  __syncthreads();  // each wave drained its own async ops above; barrier publishes LDS

  // ---- Phase 2: gather from LDS, stream out ----
  constexpr int GROUPS = NUM_LEAVES_C / 4;   // 512 float4 groups per output row
  constexpr int GITER  = GROUPS / THREADS;   // 2 groups per thread
  const v4i* __restrict__ idx4 = (const v4i*)idx;
  v4f* __restrict__ out4 = (v4f*)out;

#pragma unroll
  for (int gi = 0; gi < GITER; ++gi) {
    const int g = gi * THREADS + tid;
    const v4i iv = idx4[g];               // coalesced 16B, L2-resident, reused over 8 rows
#pragma unroll
    for (int r = 0; r < ROWS_PER_BLOCK; ++r) {
      const float* __restrict__ rowp = tile + r * NUM_INPUTS_C;
      v4f o;
      o.x = rowp[iv.x];
      o.y = rowp[iv.y];
      o.z = rowp[iv.z];
      o.w = rowp[iv.w];
      // streaming output, no reuse -> non-temporal 16B store, coalesced across lanes
      __builtin_nontemporal_store(o, &out4[(row0 + r) * GROUPS + g]);
    }
  }
}

extern "C" void kernel_launch(void* const* d_in, const int* in_sizes, int n_in,
                              void* d_out, int out_size, void* d_ws, size_t ws_size,
                              hipStream_t stream) {
  const float* x  = (const float*)d_in[0];
  const int* idx  = (const int*)d_in[1];
  float* out      = (float*)d_out;

  const int batch = in_sizes[0] / NUM_INPUTS_C;       // 16384
  dim3 grid(batch / ROWS_PER_BLOCK);                  // 2048 workgroups
  dim3 block(THREADS);
  gather_cols_kernel<<<grid, block, 0, stream>>>(x, idx, out);
}